// COPNLL_83786222010596
// MI455X (gfx1250) — compile-verified
//
#include <hip/hip_runtime.h>
#include <hip/hip_bf16.h>
#include <math.h>

#define NN 4096
#define QQ 2048
#define LL 8192
#define NB 64
#define LDP 68   // padded LDS row stride (floats): 64 data + 4 pad -> conflict-free frags

typedef __attribute__((ext_vector_type(2))) float v2f;
typedef __attribute__((ext_vector_type(8))) float v8f;
typedef __attribute__((ext_vector_type(4))) unsigned int v4u;
typedef __attribute__((ext_vector_type(4))) int v4i;
typedef __attribute__((ext_vector_type(8))) int v8i;

#ifndef __has_builtin
#define __has_builtin(x) 0
#endif
#if __has_builtin(__builtin_amdgcn_tensor_load_to_lds)
#define HAVE_TDM 1
#else
#define HAVE_TDM 0
#endif

// ---------------- Kernel 1: build R (gather + exp) -----------------
// R[i,j] = (sig2e*(i==j) + sig2bs*exp(-dist[zi[i],zi[j]]/(2*ls))) / (sig2e+sig2bs)
__global__ void build_R(const float* __restrict__ dist,
                        const long long* __restrict__ zidx,
                        const float* __restrict__ sig2e,
                        const float* __restrict__ sig2bs,
                        const float* __restrict__ ls,
                        float* __restrict__ R)
{
    long long idx = (long long)blockIdx.x * blockDim.x + threadIdx.x;
    long long total = (long long)NN * NN;
    if (idx >= total) return;
    int i = (int)(idx / NN);
    int j = (int)(idx - (long long)i * NN);
    int zi = (int)zidx[i];
    int zj = (int)zidx[j];
    float d   = dist[(long long)zi * LL + zj];
    float s2e = sig2e[0], s2b = sig2bs[0], l = ls[0];
    float v = s2b * expf(-d / (2.0f * l));
    if (i == j) v += s2e;
    R[idx] = v / (s2e + s2b);
}

// ---------------- Kernel 2: y, zs, partial sums -----------------
__global__ void compute_zs(const float* __restrict__ yt,
                           const float* __restrict__ yp,
                           const float* __restrict__ sig2e,
                           const float* __restrict__ sig2bs,
                           float* __restrict__ zs,
                           float* __restrict__ p_y,
                           float* __restrict__ p_e)
{
    __shared__ float ry[256];
    __shared__ float re[256];
    int tid = threadIdx.x;
    int i = blockIdx.x * 256 + tid;
    const float EUL = 0.57721566490153286f;        // euler_gamma
    const float Cc  = 0.77969680123362926f;        // sqrt(6)/pi
    float sig2 = sig2e[0] + sig2bs[0];
    float y = 0.0f, e = 0.0f;
    if (i < NN) {
        y = (yt[i] - yp[i]) / (Cc * sqrtf(sig2)) + EUL;
        e = expf(-y);
        float u = expf(-e);
        u = fminf(fmaxf(u, 1e-6f), 1.0f - 1e-6f);
        zs[i] = 1.41421356237309515f * erfinvf(2.0f * u - 1.0f);
    }
    ry[tid] = y; re[tid] = e;
    __syncthreads();
    for (int off = 128; off > 0; off >>= 1) {
        if (tid < off) { ry[tid] += ry[tid + off]; re[tid] += re[tid + off]; }
        __syncthreads();
    }
    if (tid == 0) { p_y[blockIdx.x] = ry[0]; p_e[blockIdx.x] = re[0]; }
}

// ---------------- Kernel 3a: factor 64x64 diagonal block in LDS -----------------
__global__ void potf2(float* __restrict__ R, int k0)
{
    __shared__ float a[NB * NB];
    int tid = threadIdx.x;  // 64 threads
    for (int idx = tid; idx < NB * NB; idx += NB)
        a[idx] = R[(long long)(k0 + idx / NB) * NN + k0 + (idx % NB)];
    __syncthreads();
    for (int j = 0; j < NB; ++j) {
        if (tid == 0) {
            float s = a[j * NB + j];
            for (int t = 0; t < j; ++t) s -= a[j * NB + t] * a[j * NB + t];
            a[j * NB + j] = sqrtf(s);
        }
        __syncthreads();
        if (tid > j) {
            float s = a[tid * NB + j];
            for (int t = 0; t < j; ++t) s -= a[tid * NB + t] * a[j * NB + t];
            a[tid * NB + j] = s / a[j * NB + j];
        }
        __syncthreads();
    }
    for (int idx = tid; idx < NB * NB; idx += NB)
        R[(long long)(k0 + idx / NB) * NN + k0 + (idx % NB)] = a[idx];
}

// ---------------- Kernel 3b: panel TRSM  X = B * L^{-T} -----------------
__global__ void trsm(float* __restrict__ R, int k0)
{
    __shared__ float Ld[NB * NB];
    int tid = threadIdx.x;  // 128 threads, one row each
    for (int idx = tid; idx < NB * NB; idx += 128)
        Ld[idx] = R[(long long)(k0 + idx / NB) * NN + k0 + (idx % NB)];
    __syncthreads();
    int r = k0 + NB + blockIdx.x * 128 + tid;
    if (r >= NN) return;
    float x[NB];
    float* row = R + (long long)r * NN + k0;
    for (int j = 0; j < NB; ++j) x[j] = row[j];
    for (int j = 0; j < NB; ++j) {
        float s = x[j];
        for (int t = 0; t < j; ++t) s -= x[t] * Ld[j * NB + t];
        x[j] = s / Ld[j * NB + j];
    }
    for (int j = 0; j < NB; ++j) row[j] = x[j];
}

// ---------------- TDM helper: DMA one (rows x 64) f32 tile into LDS -----------------
// D# per ISA ch.8: group0 = {flags, lds_addr, global_addr, type=2},
// group1 = {data_size=4B, pad 64dw+4dw -> LDS stride 68 floats, dims/strides}.
__device__ __forceinline__ void tdm_load_tile(unsigned lds_off, const float* gptr, int rows)
{
#if HAVE_TDM
    unsigned long long ga = (unsigned long long)(uintptr_t)gptr;
    v4u g0;
    g0.x = 1u;                                             // count=1, user descriptor
    g0.y = lds_off;                                        // lds_addr (bytes)
    g0.z = (unsigned)(ga & 0xFFFFFFFFull);                 // global_addr[31:0]
    g0.w = (unsigned)((ga >> 32) & 0x01FFFFFFull)          // global_addr[56:32]
         | (2u << 30);                                     // type = 2 ("image")
    v8i g1;
    g1[0] = (int)((2u << 16)      // data_size = 4 bytes
                | (1u << 20)      // pad_enable
                | (5u << 22)      // pad_interval: 64 dwords
                | (3u << 25));    // pad_amount: 4 dwords  -> stride 68 floats
    g1[1] = (int)((64u & 0xFFFFu) << 16);                  // tensor_dim0 = 64 (lo16)
    g1[2] = (int)(((unsigned)rows & 0xFFFFu) << 16);       // dim0 hi=0 | tensor_dim1 lo
    g1[3] = (int)(64u << 16);                              // dim1 hi=0 | tile_dim0 = 64
    g1[4] = rows & 0xFFFF;                                 // tile_dim1 = rows, tile_dim2=0
    g1[5] = NN;                                            // tensor_dim0_stride = 4096
    g1[6] = 0;
    g1[7] = 0;
    v4i z4 = {0, 0, 0, 0};
#if __clang_major__ >= 23
    v8i z8 = {0, 0, 0, 0, 0, 0, 0, 0};
    __builtin_amdgcn_tensor_load_to_lds(g0, g1, z4, z4, z8, 0);
#else
    __builtin_amdgcn_tensor_load_to_lds(g0, g1, z4, z4, 0);
#endif
#else
    (void)lds_off; (void)gptr; (void)rows;
#endif
}

// ---------------- Kernel 3c: LDS-blocked SYRK trailing update with fp32 WMMA ------
// Macro tile 128x64 per 256-thread block; panels staged in LDS (TDM when available);
// 8 waves each own a 32x32 register tile: 4 v8f accumulators, 4 WMMAs per K-step.
__global__ void __launch_bounds__(256) syrk_wmma(float* __restrict__ R, int k0)
{
    __shared__ float Pa[128 * LDP];   // 34 KB
    __shared__ float Pb[64 * LDP];    // 17 KB
    const int b0  = k0 + NB;
    const int rem = NN - b0;
    const int ti  = blockIdx.y;
    const int tj  = blockIdx.x;
    if (tj > 2 * ti + 1) return;               // block-uniform triangular cull
    const int rows_rem = rem - ti * 128;
    const int rows_a   = rows_rem < 128 ? rows_rem : 128;
    const int gA  = b0 + ti * 128;             // first row of P_i panel
    const int gB  = b0 + tj * 64;              // first row of P_j panel
    const int tid = threadIdx.x;

#if HAVE_TDM
    if (tid < 32) {                            // wave 0 drives the TDM
        tdm_load_tile((unsigned)(uintptr_t)Pa, &R[(long long)gA * NN + k0], rows_a);
        tdm_load_tile((unsigned)(uintptr_t)Pb, &R[(long long)gB * NN + k0], 64);
        __builtin_amdgcn_s_wait_tensorcnt(0);
    }
#else
    for (int idx = tid; idx < rows_a * 16; idx += 256) {
        int row = idx >> 4, c4 = (idx & 15) << 2;
        *(float4*)&Pa[row * LDP + c4] =
            *(const float4*)&R[(long long)(gA + row) * NN + k0 + c4];
    }
    for (int idx = tid; idx < 64 * 16; idx += 256) {
        int row = idx >> 4, c4 = (idx & 15) << 2;
        *(float4*)&Pb[row * LDP + c4] =
            *(const float4*)&R[(long long)(gB + row) * NN + k0 + c4];
    }
#endif
    __syncthreads();

    const int wave = tid >> 5;
    const int lane = tid & 31;
    const int wr   = (wave >> 1) * 32;         // wave row offset in macro tile
    const int wc   = (wave & 1) * 32;          // wave col offset in macro tile
    if (wr >= rows_a) return;                  // wave-uniform: EXEC stays all-ones
    const int m  = lane & 15;
    const int kb = (lane >> 4) * 2;

    v8f c00 = {0.f,0.f,0.f,0.f,0.f,0.f,0.f,0.f};
    v8f c01 = {0.f,0.f,0.f,0.f,0.f,0.f,0.f,0.f};
    v8f c10 = {0.f,0.f,0.f,0.f,0.f,0.f,0.f,0.f};
    v8f c11 = {0.f,0.f,0.f,0.f,0.f,0.f,0.f,0.f};

    for (int kk = 0; kk < NB; kk += 4) {
        // A fragments (16x4): lane m = M row, VGPR pair = K = kb, kb+1
        v2f a0 = *(const v2f*)&Pa[(wr +      m) * LDP + kk + kb];
        v2f a1 = *(const v2f*)&Pa[(wr + 16 + m) * LDP + kk + kb];
        // B fragments (4x16) for P_j^T: n = lane&15 indexes rows of P_j
        v2f b0v = *(const v2f*)&Pb[(wc +      m) * LDP + kk + kb];
        v2f b1v = *(const v2f*)&Pb[(wc + 16 + m) * LDP + kk + kb];
        c00 = __builtin_amdgcn_wmma_f32_16x16x4_f32(false, a0, false, b0v, (short)0, c00, false, false);
        c01 = __builtin_amdgcn_wmma_f32_16x16x4_f32(false, a0, false, b1v, (short)0, c01, false, false);
        c10 = __builtin_amdgcn_wmma_f32_16x16x4_f32(false, a1, false, b0v, (short)0, c10, false, false);
        c11 = __builtin_amdgcn_wmma_f32_16x16x4_f32(false, a1, false, b1v, (short)0, c11, false, false);
    }

    // C/D layout: VGPR v -> M = (lane>>4)*8 + v, N = lane&15
    const int orow = gA + wr + (lane >> 4) * 8;
    const int ocol = gB + wc + (lane & 15);
    for (int v = 0; v < 8; ++v) {
        R[(long long)(orow +      v) * NN + ocol]      -= c00[v];
        R[(long long)(orow +      v) * NN + ocol + 16] -= c01[v];
        R[(long long)(orow + 16 + v) * NN + ocol]      -= c10[v];
        R[(long long)(orow + 16 + v) * NN + ocol + 16] -= c11[v];
    }
}

// ---------------- Kernel 4a: forward solve  L y = b -----------------
__global__ void trisolve_fwd(const float* __restrict__ R,
                             const float* __restrict__ b,
                             float* __restrict__ x)
{
    __shared__ float xs[NN];      // 16 KB of 320 KB WGP LDS
    __shared__ float red[256];
    int tid = threadIdx.x;
    for (int i = 0; i < NN; ++i) {
        float s = 0.0f;
        const float* row = R + (long long)i * NN;
        for (int j = tid; j < i; j += 256) s += row[j] * xs[j];
        red[tid] = s; __syncthreads();
        for (int off = 128; off > 0; off >>= 1) {
            if (tid < off) red[tid] += red[tid + off];
            __syncthreads();
        }
        if (tid == 0) xs[i] = (b[i] - red[0]) / row[i];
        __syncthreads();
    }
    for (int j = tid; j < NN; j += 256) x[j] = xs[j];
}

// ---------------- Kernel 4b: backward solve  L^T w = b -----------------
__global__ void trisolve_bwd(const float* __restrict__ R,
                             const float* __restrict__ b,
                             float* __restrict__ x)
{
    __shared__ float xs[NN];
    __shared__ float red[256];
    int tid = threadIdx.x;
    for (int i = NN - 1; i >= 0; --i) {
        float s = 0.0f;
        for (int j = i + 1 + tid; j < NN; j += 256)
            s += R[(long long)j * NN + i] * xs[j];
        red[tid] = s; __syncthreads();
        for (int off = 128; off > 0; off >>= 1) {
            if (tid < off) red[tid] += red[tid + off];
            __syncthreads();
        }
        if (tid == 0) xs[i] = (b[i] - red[0]) / R[(long long)i * NN + i];
        __syncthreads();
    }
    for (int j = tid; j < NN; j += 256) x[j] = xs[j];
}

// ---------------- Kernel 5: final scalar combine -----------------
__global__ void finalize(const float* __restrict__ R,
                         const float* __restrict__ zs,
                         const float* __restrict__ w,
                         const float* __restrict__ p_y,
                         const float* __restrict__ p_e,
                         const float* __restrict__ sig2e,
                         const float* __restrict__ sig2bs,
                         float* __restrict__ out)
{
    __shared__ float r1[256], r2[256], r3[256];
    int tid = threadIdx.x;
    float sl = 0.f, sq = 0.f, sz = 0.f;
    for (int i = tid; i < NN; i += 256) {
        float d = R[(long long)i * NN + i];
        sl += logf(d);
        float z = zs[i], ww = w[i];
        sq += z * ww;
        sz += z * z;
    }
    r1[tid] = sl; r2[tid] = sq; r3[tid] = sz;
    __syncthreads();
    for (int off = 128; off > 0; off >>= 1) {
        if (tid < off) {
            r1[tid] += r1[tid + off];
            r2[tid] += r2[tid + off];
            r3[tid] += r3[tid + off];
        }
        __syncthreads();
    }
    if (tid == 0) {
        float sy = 0.f, se = 0.f;
        for (int k = 0; k < NN / 256; ++k) { sy += p_y[k]; se += p_e[k]; }
        float sig2 = sig2e[0] + sig2bs[0];
        float Nf = (float)NN;
        const float C2 = 0.60792710185402663f;   // 6/pi^2
        float mld = 2.f * sy + 2.f * se + Nf * logf(sig2) + Nf * logf(C2);
        out[0] = mld + r2[0] - r3[0] + 2.f * r1[0];
    }
}

extern "C" void kernel_launch(void* const* d_in, const int* in_sizes, int n_in,
                              void* d_out, int out_size, void* d_ws, size_t ws_size,
                              hipStream_t stream)
{
    const float*     y_true = (const float*)d_in[0];
    const float*     y_pred = (const float*)d_in[1];
    const float*     sig2e  = (const float*)d_in[2];
    const float*     sig2bs = (const float*)d_in[3];
    const float*     ls     = (const float*)d_in[4];
    const float*     dist   = (const float*)d_in[5];
    const long long* zidx   = (const long long*)d_in[6];  // reference dtype is int64
    float* out = (float*)d_out;

    // workspace layout: R (64 MB) | zs | tv | w | p_y | p_e
    float* R   = (float*)d_ws;
    float* zs  = R  + (size_t)NN * NN;
    float* tv  = zs + NN;
    float* w   = tv + NN;
    float* p_y = w  + NN;
    float* p_e = p_y + 32;

    // 1. build R (HBM-heavy gather into dist_matrix; R then lives in L2)
    {
        long long total = (long long)NN * NN;
        int blocks = (int)((total + 255) / 256);
        build_R<<<blocks, 256, 0, stream>>>(dist, zidx, sig2e, sig2bs, ls, R);
    }

    // 2. elementwise y / zs + partial sums
    compute_zs<<<NN / 256, 256, 0, stream>>>(y_true, y_pred, sig2e, sig2bs,
                                             zs, p_y, p_e);

    // 3. blocked right-looking Cholesky (fp32 WMMA trailing updates)
    for (int k0 = 0; k0 < NN; k0 += NB) {
        potf2<<<1, NB, 0, stream>>>(R, k0);
        int rem = NN - k0 - NB;
        if (rem > 0) {
            int rblocks = (rem + 127) / 128;
            trsm<<<rblocks, 128, 0, stream>>>(R, k0);
            dim3 g((rem + 63) / 64, (rem + 127) / 128);
            syrk_wmma<<<g, 256, 0, stream>>>(R, k0);
        }
    }

    // 4. cho_solve: L tv = zs ; L^T w = tv
    trisolve_fwd<<<1, 256, 0, stream>>>(R, zs, tv);
    trisolve_bwd<<<1, 256, 0, stream>>>(R, tv, w);

    // 5. combine scalar result
    finalize<<<1, 256, 0, stream>>>(R, zs, w, p_y, p_e, sig2e, sig2bs, out);
}